// MySOPBC_54451595379013
// MI455X (gfx1250) — compile-verified
//
#include <hip/hip_runtime.h>
#include <cstdint>

// ---------------- problem constants ----------------
namespace {
constexpr int BATCH   = 4;
constexpr int SEQ     = 4096;     // power of two -> mod via mask
constexpr int NM      = 2;
constexpr int LWC     = 50;
constexpr int HALF_LW = LWC / 2;  // 25
constexpr int NPAIR   = 449;      // |m*n| <= 25 pairs, verified below
constexpr int TILE    = 128;      // positions per workgroup
constexpr int WIN     = TILE + 2 * LWC;   // 228 complex values in LDS
constexpr int NWAVE   = 8;        // 256 threads / wave32
constexpr int CH1     = (NPAIR + 1) / 2;  // 225 WMMA chunks, stage 1 (1 zero pad)
constexpr float GAMMA = 0.1f;
constexpr float LOG2_10_OVER_10 = 0.3321928094887362f;

constexpr int iabs(int x) { return x < 0 ? -x : x; }

struct PairTab { int16_t m[NPAIR + 1]; int16_t n[NPAIR + 1]; };

constexpr PairTab makeTab() {
    PairTab t{};
    int k = 0;
    for (int m = -HALF_LW; m <= HALF_LW; ++m)
        for (int n = -HALF_LW; n <= HALF_LW; ++n)
            if (iabs(m * n) <= HALF_LW) { t.m[k] = (int16_t)m; t.n[k] = (int16_t)n; ++k; }
    t.m[NPAIR] = 0; t.n[NPAIR] = 0;   // zero-weight pad slot
    return t;
}
constexpr int countPairs() {
    int k = 0;
    for (int m = -HALF_LW; m <= HALF_LW; ++m)
        for (int n = -HALF_LW; n <= HALF_LW; ++n)
            if (iabs(m * n) <= HALF_LW) ++k;
    return k;
}
static_assert(countPairs() == NPAIR, "pair count mismatch vs reference _get_index()");
} // namespace

__constant__ PairTab c_tab = makeTab();

// ---------------- WMMA fragment types ----------------
typedef __attribute__((ext_vector_type(2))) float f32x2;  // A / B frag: 16x4 / 4x16 f32
typedef __attribute__((ext_vector_type(8))) float f32x8;  // C / D frag: 16x16 f32

__device__ __forceinline__ float2 cmul(float2 a, float2 b) {
    return make_float2(a.x * b.x - a.y * b.y, a.x * b.y + a.y * b.x);
}
__device__ __forceinline__ float2 cmulconj(float2 t, float2 c) { // t * conj(c)
    return make_float2(t.x * c.x + t.y * c.y, t.y * c.x - t.x * c.y);
}

// ============= Stage 1: E1 = Ps*<feat,w1>, E1m = Ps*<feat,w2> =============
// Per wave: 16 consecutive positions (matrix rows M). Each WMMA consumes two
// complex triplet terms packed as K=4 (re,im per lane-half) and accumulates
// 4 useful columns: [E1.re, E1.im, E1m.re, E1m.im]. B fragments are
// pre-permuted per (s, column) into LDS so the inner loop is a branchless
// strided ds_load walk. Unroll capped to keep the hot loop I$-resident.
__global__ __launch_bounds__(256) void sopbc_stage1(
    const float* __restrict__ Ere, const float* __restrict__ Eim,
    const float* __restrict__ task,
    const float* __restrict__ w1r, const float* __restrict__ w1i,
    const float* __restrict__ w2r, const float* __restrict__ w2i,
    float2* __restrict__ E1, float2* __restrict__ E1m)
{
    __shared__ float2 winE[WIN];
    __shared__ float2 rec4[2 * CH1 * 4];   // per s: cols 0..3 -> (b0,b1) row pairs
    __shared__ int    smn[2 * CH1];
    __shared__ float  stg[NWAVE][8][32];
    __shared__ float2 zslot;

    const int tid   = threadIdx.x;
    const int TILES = SEQ / TILE;
    const int tile  = blockIdx.x % TILES;
    const int bm    = blockIdx.x / TILES;
    const int mode  = bm % NM;
    const int b     = bm / NM;
    const int l0    = tile * TILE;

    for (int i = tid; i < WIN; i += 256) {
        int src = (l0 - LWC + i) & (SEQ - 1);          // circular, SEQ pow2
        int g = (b * SEQ + src) * NM + mode;
        winE[i] = make_float2(Ere[g], Eim[g]);
    }
    for (int s = tid; s < 2 * CH1; s += 256) {
        float ar = (s < NPAIR) ? w1r[s] : 0.f;
        float ai = (s < NPAIR) ? w1i[s] : 0.f;
        float br = (s < NPAIR) ? w2r[s] : 0.f;
        float bi = (s < NPAIR) ? w2i[s] : 0.f;
        rec4[s * 4 + 0] = make_float2(ar, -ai);   // col0: E1.re
        rec4[s * 4 + 1] = make_float2(ai,  ar);   // col1: E1.im
        rec4[s * 4 + 2] = make_float2(br, -bi);   // col2: E1m.re
        rec4[s * 4 + 3] = make_float2(bi,  br);   // col3: E1m.im
        smn[s] = (((int)c_tab.m[s]) & 0xFFFF) | (((int)c_tab.n[s]) << 16);
    }
    if (tid == 0) zslot = make_float2(0.f, 0.f);
    __syncthreads();

    const int lane   = tid & 31;
    const int wave   = tid >> 5;
    const int halfid = lane >> 4;      // lane-half selects s-term (K 0,1 vs 2,3)
    const int pidx   = lane & 15;      // matrix row M (position) == B column N
    const int ploc   = wave * 16 + pidx + LWC;

    // Loop-invariant strided LDS walkers (branchless inner loop).
    const float2* bptr = (pidx < 4) ? &rec4[halfid * 4 + pidx] : &zslot;
    const int     bstep = (pidx < 4) ? 8 : 0;      // float2 elems per chunk
    const int*    mnp  = &smn[halfid];

    f32x8 acc = {};
#pragma clang loop unroll_count(4)
    for (int c = 0; c < CH1; ++c) {
        int mn = *mnp; mnp += 2;
        int m = (int)(int16_t)(mn & 0xFFFF);
        int n = mn >> 16;
        float2 a  = winE[ploc - m];
        float2 bb = winE[ploc - n];
        float2 cc = winE[ploc - m - n];
        float2 f  = cmulconj(cmul(a, bb), cc);     // triplet feature
        float2 bv = *bptr; bptr += bstep;
        f32x2 Af; Af.x = f.x;  Af.y = f.y;         // K slots: re, im
        f32x2 Bf; Bf.x = bv.x; Bf.y = bv.y;        // rows K: b0, b1
        acc = __builtin_amdgcn_wmma_f32_16x16x4_f32(false, Af, false, Bf,
                                                    (short)0, acc, false, false);
    }

#pragma unroll
    for (int r = 0; r < 8; ++r) stg[wave][r][lane] = acc[r];
    __syncthreads();

    if (halfid == 0) {
        int p  = pidx;
        int sl = (p >> 3) * 16;    // C layout: rows M>=8 live in lanes 16..31
        int r  = p & 7;
        float Ps = exp2f(task[b * 4] * LOG2_10_OVER_10) * (1.0f / NM);
        float2 e1  = make_float2(stg[wave][r][sl + 0] * Ps, stg[wave][r][sl + 1] * Ps);
        float2 e1m = make_float2(stg[wave][r][sl + 2] * Ps, stg[wave][r][sl + 3] * Ps);
        int l = l0 + wave * 16 + p;
        int g = (b * SEQ + l) * NM + mode;
        E1[g]  = e1;
        E1m[g] = e1m;
    }
}

// ============= Stage 2: E2 from F1(E1m,E,E) + F2(E,E,E1m), output Eo =============
// Term reorder (sum is order-free): lane-half 0 processes the 449 F1 terms,
// lane-half 1 the 449 F2 terms -> pattern and base pointers are loop-invariant.
__global__ __launch_bounds__(256) void sopbc_stage2(
    const float* __restrict__ Ere, const float* __restrict__ Eim,
    const float* __restrict__ task,
    const float* __restrict__ fcr, const float* __restrict__ fci,
    const float2* __restrict__ E1, const float2* __restrict__ E1m,
    float* __restrict__ out)
{
    __shared__ float2 winE[WIN];
    __shared__ float2 winM[WIN];
    __shared__ float2 rec2[2 * NPAIR * 2];  // per t: cols 0..1 -> (b0,b1)
    __shared__ int    smn[2 * NPAIR];
    __shared__ float  stg[NWAVE][8][32];
    __shared__ float2 zslot;

    const int tid   = threadIdx.x;
    const int TILES = SEQ / TILE;
    const int tile  = blockIdx.x % TILES;
    const int bm    = blockIdx.x / TILES;
    const int mode  = bm % NM;
    const int b     = bm / NM;
    const int l0    = tile * TILE;

    for (int i = tid; i < WIN; i += 256) {
        int src = (l0 - LWC + i) & (SEQ - 1);
        int g = (b * SEQ + src) * NM + mode;
        winE[i] = make_float2(Ere[g], Eim[g]);
        winM[i] = E1m[g];
    }
    for (int t = tid; t < 2 * NPAIR; t += 256) {
        int sp = (t < NPAIR) ? t : t - NPAIR;
        float fr = fcr[t], fi = fci[t];
        rec2[t * 2 + 0] = make_float2(fr, -fi);   // col0: E2.re
        rec2[t * 2 + 1] = make_float2(fi,  fr);   // col1: E2.im
        smn[t] = (((int)c_tab.m[sp]) & 0xFFFF) | (((int)c_tab.n[sp]) << 16);
    }
    if (tid == 0) zslot = make_float2(0.f, 0.f);
    __syncthreads();

    const int lane   = tid & 31;
    const int wave   = tid >> 5;
    const int halfid = lane >> 4;
    const int pidx   = lane & 15;
    const int ploc   = wave * 16 + pidx + LWC;

    // half 0 -> F1 terms t = c        : a = E1m[l-m], conj target = E[l-m-n]
    // half 1 -> F2 terms t = NPAIR + c: a = E  [l-m], conj target = E1m[l-m-n]
    const float2* baseA = halfid ? winE : winM;
    const float2* baseC = halfid ? winM : winE;
    const float2* bptr  = (pidx < 2) ? &rec2[halfid * NPAIR * 2 + pidx] : &zslot;
    const int     bstep = (pidx < 2) ? 2 : 0;
    const int*    mnp   = &smn[halfid * NPAIR];

    f32x8 acc = {};
#pragma clang loop unroll_count(4)
    for (int c = 0; c < NPAIR; ++c) {
        int mn = *mnp; mnp += 1;
        int m = (int)(int16_t)(mn & 0xFFFF);
        int n = mn >> 16;
        float2 a  = baseA[ploc - m];
        float2 bb = winE [ploc - n];
        float2 cc = baseC[ploc - m - n];
        float2 f  = cmulconj(cmul(a, bb), cc);
        float2 bv = *bptr; bptr += bstep;
        f32x2 Af; Af.x = f.x;  Af.y = f.y;
        f32x2 Bf; Bf.x = bv.x; Bf.y = bv.y;
        acc = __builtin_amdgcn_wmma_f32_16x16x4_f32(false, Af, false, Bf,
                                                    (short)0, acc, false, false);
    }

#pragma unroll
    for (int r = 0; r < 8; ++r) stg[wave][r][lane] = acc[r];
    __syncthreads();

    if (halfid == 0) {
        int p = pidx;
        int l = l0 + wave * 16 + p;
        if (l >= LWC && l < SEQ - LWC) {
            int sl = (p >> 3) * 16;
            int r  = p & 7;
            float Ps = exp2f(task[b * 4] * LOG2_10_OVER_10) * (1.0f / NM);
            float e2r = stg[wave][r][sl + 0] * Ps;
            float e2i = stg[wave][r][sl + 1] * Ps;
            int g = (b * SEQ + l) * NM + mode;
            float2 e  = winE[wave * 16 + p + LWC];
            float2 e1 = E1[g];
            float eor = e.x + GAMMA * e1.x + GAMMA * GAMMA * e2r;
            float eoi = e.y + GAMMA * e1.y + GAMMA * GAMMA * e2i;
            int o = ((b * (SEQ - 2 * LWC) + (l - LWC)) * NM + mode) * 2;
            out[o + 0] = eor;
            out[o + 1] = eoi;
        }
    }
}

// ---------------- launcher ----------------
extern "C" void kernel_launch(void* const* d_in, const int* in_sizes, int n_in,
                              void* d_out, int out_size, void* d_ws, size_t ws_size,
                              hipStream_t stream)
{
    const float* Ere  = (const float*)d_in[0];
    const float* Eim  = (const float*)d_in[1];
    const float* task = (const float*)d_in[2];
    const float* w1r  = (const float*)d_in[3];
    const float* w1i  = (const float*)d_in[4];
    const float* w2r  = (const float*)d_in[5];
    const float* w2i  = (const float*)d_in[6];
    const float* fcr  = (const float*)d_in[7];
    const float* fci  = (const float*)d_in[8];

    float2* E1  = (float2*)d_ws;                       // 256 KB
    float2* E1m = E1 + (size_t)BATCH * SEQ * NM;       // 256 KB

    dim3 grid(BATCH * NM * (SEQ / TILE));              // 256 blocks
    dim3 blk(256);                                     // 8 wave32
    sopbc_stage1<<<grid, blk, 0, stream>>>(Ere, Eim, task, w1r, w1i, w2r, w2i, E1, E1m);
    sopbc_stage2<<<grid, blk, 0, stream>>>(Ere, Eim, task, fcr, fci, E1, E1m, (float*)d_out);
}